// GraphSAGE_59373627900056
// MI455X (gfx1250) — compile-verified
//
#include <hip/hip_runtime.h>

typedef float v2f __attribute__((ext_vector_type(2)));
typedef float v8f __attribute__((ext_vector_type(8)));

#define N_NODES 100000
#define N_EDGES 1250000
#define D_HID 64
#define N_CLS 40

// ------------------------------------------------------------------
// Edge scatter: one wave (32 lanes) per edge; lane handles 2 features.
// agg[dst] += h[src]; deg[dst] += 1 (layer 1 only).
// ------------------------------------------------------------------
__global__ void sage_scatter(const float* __restrict__ h,
                             const long long* __restrict__ src,
                             const long long* __restrict__ dst,
                             float* __restrict__ agg,
                             float* __restrict__ deg,
                             int n_edges)
{
    int e = blockIdx.x * 8 + (threadIdx.x >> 5);
    int lane = threadIdx.x & 31;
    if (e >= n_edges) return;
    long long s = src[e];
    long long d = dst[e];
    float2 v = ((const float2*)(h + s * 64))[lane];
    float* out = agg + d * 64 + 2 * lane;
    __hip_atomic_fetch_add(out,     v.x, __ATOMIC_RELAXED, __HIP_MEMORY_SCOPE_AGENT);
    __hip_atomic_fetch_add(out + 1, v.y, __ATOMIC_RELAXED, __HIP_MEMORY_SCOPE_AGENT);
    if (deg != nullptr && lane == 0)
        __hip_atomic_fetch_add(deg + d, 1.0f, __ATOMIC_RELAXED, __HIP_MEMORY_SCOPE_AGENT);
}

// agg[i] /= max(deg[row], 1)
__global__ void sage_normalize(float* __restrict__ agg,
                               const float* __restrict__ deg,
                               int total)
{
    int i = blockIdx.x * blockDim.x + threadIdx.x;
    if (i >= total) return;
    agg[i] = agg[i] / fmaxf(deg[i >> 6], 1.0f);
}

// ------------------------------------------------------------------
// Layer-1 fused GEMM: out = relu([aggn|xin] @ [Wl;Wr] + b), K=128, ncols=64.
// Block = 128 threads = 4 waves; wave w owns n-tile [16w, 16w+16).
// A 16x4 f32 frag: lane m=lane&15, kh=lane>>4 holds (m, 2kh) / (m, 2kh+1).
// B 4x16 frag: lane n=lane&15 holds (2kh, n) / (2kh+1, n).
// D 16x16:     vgpr r, lane -> row r+8*kh, col lane&15.
// ------------------------------------------------------------------
__global__ __launch_bounds__(128) void sage_gemm64_relu(
    const float* __restrict__ aggn, const float* __restrict__ xin,
    const float* __restrict__ Wl,  const float* __restrict__ Wr,
    const float* __restrict__ bias, float* __restrict__ out)
{
    const int wave = threadIdx.x >> 5;
    const int lane = threadIdx.x & 31;
    const int m    = lane & 15;
    const int kh   = lane >> 4;              // 0 or 1
    const int row0 = blockIdx.x * 16;
    const int n    = wave * 16 + m;          // this lane's output column

    v8f acc = {};
    #pragma unroll
    for (int k0 = 0; k0 < 128; k0 += 4) {
        const int k = (k0 & 63) + 2 * kh;    // column within the selected source
        const float* Asrc = (k0 < 64) ? aggn : xin;
        const float* Bsrc = (k0 < 64) ? Wl   : Wr;
        v2f a = *(const v2f*)(Asrc + (size_t)(row0 + m) * 64 + k);
        v2f b;
        b.x = Bsrc[(k    ) * 64 + n];
        b.y = Bsrc[(k + 1) * 64 + n];
        acc = __builtin_amdgcn_wmma_f32_16x16x4_f32(false, a, false, b,
                                                    (short)0, acc, false, false);
    }
    const float bn = bias[n];
    #pragma unroll
    for (int r = 0; r < 8; ++r) {
        int mr = row0 + r + 8 * kh;
        out[(size_t)mr * 64 + n] = fmaxf(acc[r] + bn, 0.0f);
    }
}

// ------------------------------------------------------------------
// Layer-2 fused GEMM: out = [aggn|hin] @ [Wl;Wr] + b, K=128, ncols=40
// padded to 48 (3 waves); cols >= 40 get zero B so EXEC stays all-1s
// for WMMA; their results are simply not stored.
// ------------------------------------------------------------------
__global__ __launch_bounds__(96) void sage_gemm40(
    const float* __restrict__ aggn, const float* __restrict__ hin,
    const float* __restrict__ Wl,  const float* __restrict__ Wr,
    const float* __restrict__ bias, float* __restrict__ out)
{
    const int wave = threadIdx.x >> 5;       // 0..2
    const int lane = threadIdx.x & 31;
    const int m    = lane & 15;
    const int kh   = lane >> 4;
    const int row0 = blockIdx.x * 16;
    const int n    = wave * 16 + m;
    const bool nv  = (n < N_CLS);

    v8f acc = {};
    #pragma unroll
    for (int k0 = 0; k0 < 128; k0 += 4) {
        const int k = (k0 & 63) + 2 * kh;
        const float* Asrc = (k0 < 64) ? aggn : hin;
        const float* Bsrc = (k0 < 64) ? Wl   : Wr;
        v2f a = *(const v2f*)(Asrc + (size_t)(row0 + m) * 64 + k);
        v2f b;
        b.x = nv ? Bsrc[(k    ) * N_CLS + n] : 0.0f;
        b.y = nv ? Bsrc[(k + 1) * N_CLS + n] : 0.0f;
        acc = __builtin_amdgcn_wmma_f32_16x16x4_f32(false, a, false, b,
                                                    (short)0, acc, false, false);
    }
    if (nv) {
        const float bn = bias[n];
        #pragma unroll
        for (int r = 0; r < 8; ++r) {
            int mr = row0 + r + 8 * kh;
            out[(size_t)mr * N_CLS + n] = acc[r] + bn;
        }
    }
}

// Row-wise log_softmax over 40 columns, in place. One thread per node;
// 160B contiguous per row, trivially bandwidth-bound.
__global__ void sage_log_softmax(float* __restrict__ out, int n_nodes)
{
    int i = blockIdx.x * blockDim.x + threadIdx.x;
    if (i >= n_nodes) return;
    float* row = out + (size_t)i * N_CLS;
    float v[N_CLS];
    float mx = -__builtin_inff();
    #pragma unroll
    for (int j = 0; j < N_CLS; ++j) { v[j] = row[j]; mx = fmaxf(mx, v[j]); }
    float s = 0.0f;
    #pragma unroll
    for (int j = 0; j < N_CLS; ++j) s += __expf(v[j] - mx);
    float lse = mx + __logf(s);
    #pragma unroll
    for (int j = 0; j < N_CLS; ++j) row[j] = v[j] - lse;
}

extern "C" void kernel_launch(void* const* d_in, const int* in_sizes, int n_in,
                              void* d_out, int out_size, void* d_ws, size_t ws_size,
                              hipStream_t stream)
{
    const float*     x   = (const float*)d_in[0];
    const long long* ei  = (const long long*)d_in[1];   // int64 edge_index [2, E]
    const float*     W1l = (const float*)d_in[2];
    const float*     b1  = (const float*)d_in[3];
    const float*     W1r = (const float*)d_in[4];
    const float*     W2l = (const float*)d_in[5];
    const float*     b2  = (const float*)d_in[6];
    const float*     W2r = (const float*)d_in[7];
    float* out = (float*)d_out;

    // workspace: agg [N,64] | deg [N] | h1 [N,64]   (~52 MB)
    float* agg = (float*)d_ws;
    float* deg = agg + (size_t)N_NODES * 64;
    float* h1  = deg + N_NODES;

    const long long* src = ei;
    const long long* dst = ei + N_EDGES;

    const int sc_blocks = (N_EDGES + 7) / 8;         // 8 edges per 256-thread block
    const int tot       = N_NODES * 64;
    const int gemm_blocks = N_NODES / 16;            // 100000 % 16 == 0

    // ---- layer 1 ----
    hipMemsetAsync(agg, 0, (size_t)N_NODES * 64 * sizeof(float), stream);
    hipMemsetAsync(deg, 0, (size_t)N_NODES * sizeof(float), stream);
    sage_scatter<<<sc_blocks, 256, 0, stream>>>(x, src, dst, agg, deg, N_EDGES);
    sage_normalize<<<(tot + 255) / 256, 256, 0, stream>>>(agg, deg, tot);
    sage_gemm64_relu<<<gemm_blocks, 128, 0, stream>>>(agg, x, W1l, W1r, b1, h1);

    // ---- layer 2 ----
    hipMemsetAsync(agg, 0, (size_t)N_NODES * 64 * sizeof(float), stream);
    sage_scatter<<<sc_blocks, 256, 0, stream>>>(h1, src, dst, agg, nullptr, N_EDGES);
    sage_normalize<<<(tot + 255) / 256, 256, 0, stream>>>(agg, deg, tot);
    sage_gemm40<<<gemm_blocks, 96, 0, stream>>>(agg, h1, W2l, W2r, b2, out);

    // ---- log_softmax ----
    sage_log_softmax<<<(N_NODES + 255) / 256, 256, 0, stream>>>(out, N_NODES);
}